// GATModel_11081015623742
// MI455X (gfx1250) — compile-verified
//
#include <hip/hip_runtime.h>
#include <hip/hip_bf16.h>

#define LRELU_SLOPE 0.2f
#define GAT_EPS 1e-16f
#define KT 32   // k-tile staged in LDS

typedef __attribute__((ext_vector_type(2))) float v2f;
typedef __attribute__((ext_vector_type(8))) float v8f;

// ---------------------------------------------------------------------------
// Utility: fill a float buffer
// ---------------------------------------------------------------------------
__global__ void fill_kernel(float* __restrict__ p, float v, int n) {
    int i = blockIdx.x * blockDim.x + threadIdx.x;
    if (i < n) p[i] = v;
}

// ---------------------------------------------------------------------------
// GEMM: C[M,64] = A[M,K] x B[K,64]   (fp32 WMMA 16x16x4, wave32)
//
// Block = 128 threads = 4 waves, owns a 32-row x 64-col output tile.
// Wave w owns columns w*16..w*16+15 with TWO accumulators (rows 0-15 and
// 16-31 of the tile) so each B fragment load feeds two WMMAs.
// The 32xKT A tile is staged once per block in LDS (padded row stride 33
// floats -> conflict-free on 64 banks) and shared by all 4 waves.
//
// fp32 A 16x4 layout (ISA 7.12.2): lanes 0-15 = rows, VGPR0 = {K0|K2},
// VGPR1 = {K1|K3} split at lane 16.  B 4x16 mirrors it.  C/D: VGPR r holds
// row r (lanes 0-15) / row r+8 (lanes 16-31), col = lane&15.
// EXEC is all-ones at every WMMA (guards only on stores).
// ---------------------------------------------------------------------------
__global__ void gemm_wmma_f32(const float* __restrict__ A,
                              const float* __restrict__ B,
                              float* __restrict__ C, int M, int K) {
    __shared__ float As[32][KT + 1];

    const int tid  = threadIdx.x;
    const int lane = tid & 31;
    const int wave = tid >> 5;              // 0..3 -> column tile
    const int r0   = blockIdx.x * 32;
    const int c0   = wave * 16;
    const int half = lane >> 4;             // 0 or 1
    const int l15  = lane & 15;
    const int bcol = c0 + l15;

    v8f acc0 = {}, acc1 = {};

    for (int k0 = 0; k0 < K; k0 += KT) {
        __syncthreads();                    // protect previous-tile reads
        // cooperative stage: 32 rows x KT cols = 1024 floats, 8 per thread
#pragma unroll
        for (int i = tid; i < 32 * KT; i += 128) {
            const int r = i >> 5;           // KT == 32
            const int c = i & (KT - 1);
            const int grow = r0 + r;
            As[r][c] = (grow < M) ? A[(long long)grow * K + k0 + c] : 0.f;
        }
        __syncthreads();

#pragma unroll
        for (int k = 0; k < KT; k += 4) {
            const int kk = k + (half << 1);
            v2f b, a0, a1;
            b.x  = B[(k0 + kk) * 64 + bcol];
            b.y  = B[(k0 + kk + 1) * 64 + bcol];
            a0.x = As[l15][kk];       a0.y = As[l15][kk + 1];
            a1.x = As[16 + l15][kk];  a1.y = As[16 + l15][kk + 1];
            // 8 args: (neg_a, A, neg_b, B, c_mod, C, reuse_a, reuse_b)
            acc0 = __builtin_amdgcn_wmma_f32_16x16x4_f32(
                false, a0, false, b, (short)0, acc0, false, false);
            acc1 = __builtin_amdgcn_wmma_f32_16x16x4_f32(
                false, a1, false, b, (short)0, acc1, false, false);
        }
    }

    const int rbase = r0 + half * 8;
#pragma unroll
    for (int r = 0; r < 8; ++r) {
        const int row0 = rbase + r;
        const int row1 = row0 + 16;
        if (row0 < M) C[(long long)row0 * 64 + c0 + l15] = acc0[r];
        if (row1 < M) C[(long long)row1 * 64 + c0 + l15] = acc1[r];
    }
}

// ---------------------------------------------------------------------------
// Per-node attention logits: asrc[n,h] = sum_c xl[n,h*CH+c]*a_src[h,c]
// ---------------------------------------------------------------------------
template <int H, int CH>
__global__ void alpha_kernel(const float* __restrict__ xl,
                             const float* __restrict__ a_src,
                             const float* __restrict__ a_dst,
                             float* __restrict__ asrc,
                             float* __restrict__ adst, int n_nodes) {
    int idx = blockIdx.x * blockDim.x + threadIdx.x;   // over n_nodes*H
    if (idx >= n_nodes * H) return;
    const int n = idx / H, h = idx % H;
    const float* row = xl + (long long)n * 64 + h * CH;
    float s1 = 0.f, s2 = 0.f;
#pragma unroll
    for (int c = 0; c < CH; ++c) {
        s1 += row[c] * a_src[h * CH + c];
        s2 += row[c] * a_dst[h * CH + c];
    }
    asrc[idx] = s1;
    adst[idx] = s2;
}

__device__ __forceinline__ float lrelu(float x) {
    return x > 0.f ? x : LRELU_SLOPE * x;
}

// float atomic max via int-max / uint-min bit trick (native atomics, no CAS)
__device__ __forceinline__ void atomic_max_f32(float* addr, float val) {
    if (val >= 0.f)
        atomicMax((int*)addr, __float_as_int(val));
    else
        atomicMin((unsigned int*)addr, __float_as_uint(val));
}

template <int H>
__device__ __forceinline__ void edge_sd(const int* __restrict__ src,
                                        const int* __restrict__ dst,
                                        int E, int e, int& s, int& d) {
    if (e < E) { s = src[e]; d = dst[e]; }
    else       { s = d = e - E; }          // self loop
}

// ---------------------------------------------------------------------------
// Edge pass 1: segment max of leaky_relu(asrc[src]+adst[dst]) into amax[dst]
// one thread per (edge, head)
// ---------------------------------------------------------------------------
template <int H>
__global__ void edge_max_kernel(const int* __restrict__ src,
                                const int* __restrict__ dst, int E, int ET,
                                const float* __restrict__ asrc,
                                const float* __restrict__ adst,
                                float* __restrict__ amax) {
    int t = blockIdx.x * blockDim.x + threadIdx.x;
    if (t >= ET * H) return;
    const int e = t / H, h = t % H;
    int s, d; edge_sd<H>(src, dst, E, e, s, d);
    float v = lrelu(asrc[s * H + h] + adst[d * H + h]);
    atomic_max_f32(&amax[d * H + h], v);
}

// ---------------------------------------------------------------------------
// Edge pass 2: denom[dst] += exp(logit - amax[dst])
// ---------------------------------------------------------------------------
template <int H>
__global__ void edge_sum_kernel(const int* __restrict__ src,
                                const int* __restrict__ dst, int E, int ET,
                                const float* __restrict__ asrc,
                                const float* __restrict__ adst,
                                const float* __restrict__ amax,
                                float* __restrict__ den) {
    int t = blockIdx.x * blockDim.x + threadIdx.x;
    if (t >= ET * H) return;
    const int e = t / H, h = t % H;
    int s, d; edge_sd<H>(src, dst, E, e, s, d);
    float v = lrelu(asrc[s * H + h] + adst[d * H + h]);
    atomicAdd(&den[d * H + h], expf(v - amax[d * H + h]));
}

// ---------------------------------------------------------------------------
// Edge pass 3: agg[dst, f] += xl[src, f] * softmax_weight(edge, head(f))
// one thread per (edge, feature); 64 features per edge; CSH = log2(ch)
// ---------------------------------------------------------------------------
template <int H, int CSH>
__global__ void edge_agg_kernel(const int* __restrict__ src,
                                const int* __restrict__ dst, int E, int ET,
                                const float* __restrict__ asrc,
                                const float* __restrict__ adst,
                                const float* __restrict__ amax,
                                const float* __restrict__ den,
                                const float* __restrict__ xl,
                                float* __restrict__ agg) {
    int t = blockIdx.x * blockDim.x + threadIdx.x;
    if (t >= ET * 64) return;
    const int e = t >> 6, f = t & 63, h = f >> CSH;
    int s, d; edge_sd<H>(src, dst, E, e, s, d);
    float v = lrelu(asrc[s * H + h] + adst[d * H + h]);
    float w = expf(v - amax[d * H + h]) / fmaxf(den[d * H + h], GAT_EPS);
    atomicAdd(&agg[(long long)d * 64 + f], xl[(long long)s * 64 + f] * w);
}

// ---------------------------------------------------------------------------
// Layer-1 epilogue: h = elu(agg + b1)
// ---------------------------------------------------------------------------
__global__ void elu_bias_kernel(const float* __restrict__ agg,
                                const float* __restrict__ bias,
                                float* __restrict__ h, int n) {
    int i = blockIdx.x * blockDim.x + threadIdx.x;
    if (i >= n) return;
    float v = agg[i] + bias[i & 63];
    h[i] = v > 0.f ? v : (expf(v) - 1.f);
}

// ---------------------------------------------------------------------------
// Layer-2 epilogue: out[n,c] = mean_h agg2[n,h,c] + b2[c]   (4 heads, 16 ch)
// ---------------------------------------------------------------------------
__global__ void final_kernel(const float* __restrict__ agg2,
                             const float* __restrict__ b2,
                             float* __restrict__ out, int n_nodes) {
    int i = blockIdx.x * blockDim.x + threadIdx.x;
    if (i >= n_nodes * 16) return;
    const int n = i >> 4, c = i & 15;
    const float* row = agg2 + (long long)n * 64;
    float s = row[c] + row[16 + c] + row[32 + c] + row[48 + c];
    out[i] = 0.25f * s + b2[c];
}

// ---------------------------------------------------------------------------
static inline int cdiv(long long a, long long b) { return (int)((a + b - 1) / b); }

extern "C" void kernel_launch(void* const* d_in, const int* in_sizes, int n_in,
                              void* d_out, int out_size, void* d_ws, size_t ws_size,
                              hipStream_t stream) {
    const float* x   = (const float*)d_in[0];
    const int*   ei  = (const int*)d_in[1];
    const float* W1  = (const float*)d_in[2];
    const float* as1 = (const float*)d_in[3];
    const float* ad1 = (const float*)d_in[4];
    const float* b1  = (const float*)d_in[5];
    const float* W2  = (const float*)d_in[6];
    const float* as2 = (const float*)d_in[7];
    const float* ad2 = (const float*)d_in[8];
    const float* b2  = (const float*)d_in[9];
    float*       out = (float*)d_out;

    const int IN = 256;
    const int n  = in_sizes[0] / IN;       // 50000
    const int E  = in_sizes[1] / 2;        // 1,600,000
    const int ET = E + n;                  // + self loops
    const int* src = ei;
    const int* dst = ei + E;

    // workspace layout (floats)
    float* ws    = (float*)d_ws;
    float* xl1   = ws;                           // n*64
    float* asrc1 = xl1   + (long long)n * 64;    // n*8
    float* adst1 = asrc1 + (long long)n * 8;     // n*8
    float* amax1 = adst1 + (long long)n * 8;     // n*8
    float* den1  = amax1 + (long long)n * 8;     // n*8
    float* agg1  = den1  + (long long)n * 8;     // n*64
    float* hbuf  = agg1  + (long long)n * 64;    // n*64
    float* xl2   = hbuf  + (long long)n * 64;    // n*64
    float* asrc2 = xl2   + (long long)n * 64;    // n*4
    float* adst2 = asrc2 + (long long)n * 4;     // n*4
    float* amax2 = adst2 + (long long)n * 4;     // n*4
    float* den2  = amax2 + (long long)n * 4;     // n*4
    float* agg2  = den2  + (long long)n * 4;     // n*64

    const int B = 256;
    const float NEG_INF = -__builtin_inff();

    // ---- init (re-run every call: deterministic under graph replay) ----
    fill_kernel<<<cdiv((long long)n * 8, B), B, 0, stream>>>(amax1, NEG_INF, n * 8);
    fill_kernel<<<cdiv((long long)n * 8, B), B, 0, stream>>>(den1, 0.f, n * 8);
    fill_kernel<<<cdiv((long long)n * 64, B), B, 0, stream>>>(agg1, 0.f, n * 64);
    fill_kernel<<<cdiv((long long)n * 4, B), B, 0, stream>>>(amax2, NEG_INF, n * 4);
    fill_kernel<<<cdiv((long long)n * 4, B), B, 0, stream>>>(den2, 0.f, n * 4);
    fill_kernel<<<cdiv((long long)n * 64, B), B, 0, stream>>>(agg2, 0.f, n * 64);

    // ---- layer 1 ----
    gemm_wmma_f32<<<cdiv(n, 32), 128, 0, stream>>>(x, W1, xl1, n, IN);
    alpha_kernel<8, 8><<<cdiv((long long)n * 8, B), B, 0, stream>>>(
        xl1, as1, ad1, asrc1, adst1, n);
    edge_max_kernel<8><<<cdiv((long long)ET * 8, B), B, 0, stream>>>(
        src, dst, E, ET, asrc1, adst1, amax1);
    edge_sum_kernel<8><<<cdiv((long long)ET * 8, B), B, 0, stream>>>(
        src, dst, E, ET, asrc1, adst1, amax1, den1);
    edge_agg_kernel<8, 3><<<cdiv((long long)ET * 64, B), B, 0, stream>>>(
        src, dst, E, ET, asrc1, adst1, amax1, den1, xl1, agg1);
    elu_bias_kernel<<<cdiv((long long)n * 64, B), B, 0, stream>>>(
        agg1, b1, hbuf, n * 64);

    // ---- layer 2 ----
    gemm_wmma_f32<<<cdiv(n, 32), 128, 0, stream>>>(hbuf, W2, xl2, n, 64);
    alpha_kernel<4, 16><<<cdiv((long long)n * 4, B), B, 0, stream>>>(
        xl2, as2, ad2, asrc2, adst2, n);
    edge_max_kernel<4><<<cdiv((long long)ET * 4, B), B, 0, stream>>>(
        src, dst, E, ET, asrc2, adst2, amax2);
    edge_sum_kernel<4><<<cdiv((long long)ET * 4, B), B, 0, stream>>>(
        src, dst, E, ET, asrc2, adst2, amax2, den2);
    edge_agg_kernel<4, 4><<<cdiv((long long)ET * 64, B), B, 0, stream>>>(
        src, dst, E, ET, asrc2, adst2, amax2, den2, xl2, agg2);
    final_kernel<<<cdiv((long long)n * 16, B), B, 0, stream>>>(agg2, b2, out, n);
}